// QLIFLayer_472446402757
// MI455X (gfx1250) — compile-verified
//
#include <hip/hip_runtime.h>

// LIF hyperparameters (match reference)
#define DECAY_F       0.95f   // 1 - dt/tau
#define GAIN_F        0.05f   // dt/tau
#define VTH_F         1.0f
#define ADAPT_DECAY_F 0.95f
#define ADAPT_INC_F   0.1f

typedef float v4f __attribute__((ext_vector_type(4)));

// Async global->LDS copy of 16 B per lane (ASYNCcnt-tracked CDNA5 path).
// s_wait_dscnt 0 first: async LDS writes are unordered vs. in-flight ds_load,
// and we re-target a buffer whose ds_load was just issued (WAR hazard).
// th:TH_LOAD_NT — the input stream is touched exactly once.
__device__ __forceinline__ void lif_issue_prefetch(const v4f* gsrc, unsigned lds_addr) {
  unsigned long long ga = (unsigned long long)(const void*)gsrc;
  asm volatile(
      "s_wait_dscnt 0x0\n\t"
      "global_load_async_to_lds_b128 %0, %1, off th:TH_LOAD_NT"
      :
      : "v"(lds_addr), "v"(ga)
      : "memory");
}

__device__ __forceinline__ void lif_step(v4f I, v4f& V, v4f& b, v4f& s_out, v4f& v_out) {
#pragma unroll
  for (int i = 0; i < 4; ++i) {
    float v    = fmaf(DECAY_F, V[i], fmaf(GAIN_F, I[i], -b[i]));
    bool  fire = v >= VTH_F;
    float sp   = fire ? 1.0f : 0.0f;
    float vp   = fire ? 0.0f : v;
    b[i]       = fmaf(ADAPT_DECAY_F, b[i], ADAPT_INC_F * sp);
    V[i]       = vp;
    s_out[i]   = sp;
    v_out[i]   = vp;
  }
}

// Requires T % 4 == 0 and T >= 8 (harness: T = 256).
__global__ __launch_bounds__(128) void lif_scan_kernel(
    const float* __restrict__ in, float* __restrict__ spk_out,
    float* __restrict__ vol_out, int T, int bn4) {
  // 4-deep LDS pipeline: each wave prefetches only slots its own lanes read,
  // so the only synchronization needed is the wave-private ASYNCcnt.
  __shared__ v4f tile[4][128];
  constexpr unsigned kBufBytes = 128 * sizeof(v4f);  // 2048 B per buffer

  const int tid = threadIdx.x;
  const int col = blockIdx.x * 128 + tid;   // float4 column, exact fit

  const size_t row = (size_t)bn4;           // timestep stride in v4f units
  const v4f* gin  = reinterpret_cast<const v4f*>(in) + col;
  v4f*       ps   = reinterpret_cast<v4f*>(spk_out) + col;
  v4f*       pv   = reinterpret_cast<v4f*>(vol_out) + col;

  // Raw LDS byte address of this lane's slot in buffer 0 (flat addr low 32
  // bits == wave-relative LDS offset on CDNA5).
  const unsigned lds0 = (unsigned)(size_t)(void*)&tile[0][tid];

  // Prologue: fill the pipeline, 4 timesteps in flight.
  const v4f* gpre = gin;
#pragma unroll
  for (int p = 0; p < 4; ++p) {
    lif_issue_prefetch(gpre, lds0 + (unsigned)p * kBufBytes);
    gpre += row;
  }

  v4f V = {0.f, 0.f, 0.f, 0.f};
  v4f b = {0.f, 0.f, 0.f, 0.f};

  // Steady state: branch-free 4-phase body; always 4 loads in flight.
  for (int t = 0; t <= T - 8; t += 4) {
#pragma unroll
    for (int p = 0; p < 4; ++p) {
      // In flight: suffix of {t+p .. t+p+3}; asynccnt<=3 => load t+p landed.
      asm volatile("s_wait_asynccnt 0x3" ::: "memory");
      v4f I = tile[p][tid];
      lif_issue_prefetch(gpre, lds0 + (unsigned)p * kBufBytes);
      gpre += row;
      v4f s, vp;
      lif_step(I, V, b, s, vp);
      __builtin_nontemporal_store(s,  ps);  ps += row;  // write-once: TH=NT
      __builtin_nontemporal_store(vp, pv);  pv += row;
    }
  }

  // Epilogue: drain last 4 timesteps, waits 3,2,1,0 (no more issues).
  {
    v4f s, vp;
    asm volatile("s_wait_asynccnt 0x3" ::: "memory");
    lif_step(tile[0][tid], V, b, s, vp);
    __builtin_nontemporal_store(s, ps);  __builtin_nontemporal_store(vp, pv);
    ps += row; pv += row;

    asm volatile("s_wait_asynccnt 0x2" ::: "memory");
    lif_step(tile[1][tid], V, b, s, vp);
    __builtin_nontemporal_store(s, ps);  __builtin_nontemporal_store(vp, pv);
    ps += row; pv += row;

    asm volatile("s_wait_asynccnt 0x1" ::: "memory");
    lif_step(tile[2][tid], V, b, s, vp);
    __builtin_nontemporal_store(s, ps);  __builtin_nontemporal_store(vp, pv);
    ps += row; pv += row;

    asm volatile("s_wait_asynccnt 0x0" ::: "memory");
    lif_step(tile[3][tid], V, b, s, vp);
    __builtin_nontemporal_store(s, ps);  __builtin_nontemporal_store(vp, pv);
  }
}

extern "C" void kernel_launch(void* const* d_in, const int* in_sizes, int n_in,
                              void* d_out, int out_size, void* d_ws, size_t ws_size,
                              hipStream_t stream) {
  const float* in  = (const float*)d_in[0];
  float*       out = (float*)d_out;

  const int BN    = 32 * 4096;          // B * N neurons per timestep
  const int total = in_sizes[0];        // T * B * N
  const int T     = total / BN;         // 256 (multiple of 4)
  const int bn4   = BN / 4;             // 32768 float4 columns

  float* spk = out;                     // spike trace first (as float 0/1)
  float* vol = out + (size_t)T * BN;    // then voltage trace

  dim3 block(128);                      // 4 wave32s
  dim3 grid(bn4 / 128);                 // 256 workgroups, exact cover
  lif_scan_kernel<<<grid, block, 0, stream>>>(in, spk, vol, T, bn4);
}